// BBoxDetector_81587198754956
// MI455X (gfx1250) — compile-verified
//
#include <hip/hip_runtime.h>
#include <math.h>

typedef float v2f __attribute__((ext_vector_type(2)));
typedef float v8f __attribute__((ext_vector_type(8)));

// 1/sqrt(1+1e-5), replaces the per-element IEEE divide (within 1 ulp of ref).
#define INV_SQ (1.0f / sqrtf(1.0f + 1e-5f))

// ---------------------------------------------------------------------------
// Farthest point sampling: one block per batch. Matches jax.lax.scan semantics:
// idx[0]=0; each step records 'far', updates min-dist with center 'far',
// next far = argmax(dist) with lowest-index tie-break. We only need new_xyz.
// ---------------------------------------------------------------------------
__global__ void __launch_bounds__(256) fps_kernel(const float* __restrict__ src,
                                                  float* __restrict__ new_xyz,
                                                  int N, int S) {
#pragma clang fp contract(off)
  __shared__ float dist[1024];
  __shared__ float rv[256];
  __shared__ int   ri[256];
  __shared__ int   sfar;
  const int b = blockIdx.x;
  const int tid = threadIdx.x;
  const float* X = src + (size_t)b * N * 3;
  for (int i = tid; i < N; i += 256) dist[i] = 1e10f;
  if (tid == 0) sfar = 0;
  __syncthreads();
  for (int k = 0; k < S; ++k) {
    const int far = sfar;
    if (tid < 3) new_xyz[((size_t)b * S + k) * 3 + tid] = X[(size_t)far * 3 + tid];
    const float cx = X[far * 3 + 0], cy = X[far * 3 + 1], cz = X[far * 3 + 2];
    float bv = -1.0f; int bi = 0;
    for (int i = tid; i < N; i += 256) {
      float dx = X[i * 3 + 0] - cx;
      float dy = X[i * 3 + 1] - cy;
      float dz = X[i * 3 + 2] - cz;
      float d = (dx * dx + dy * dy) + dz * dz;
      float nd = fminf(dist[i], d);
      dist[i] = nd;
      if (nd > bv) { bv = nd; bi = i; }
    }
    rv[tid] = bv; ri[tid] = bi;
    __syncthreads();
    for (int st = 128; st > 0; st >>= 1) {
      if (tid < st) {
        float ov = rv[tid + st]; int oi = ri[tid + st];
        if (ov > rv[tid] || (ov == rv[tid] && oi < ri[tid])) { rv[tid] = ov; ri[tid] = oi; }
      }
      __syncthreads();
    }
    if (tid == 0) sfar = ri[0];
    __syncthreads();
  }
}

// ---------------------------------------------------------------------------
// Ball query: one thread per query point. Equivalent to
// sort(where(sqdist>r2, N, arange))[:ns] with first-index fill / N-1 fallback.
// sqdist computed with the same (|a|^2 - 2ab) + |b|^2 ordering as the reference.
// ---------------------------------------------------------------------------
__global__ void __launch_bounds__(256) ball_query_kernel(
    const float* __restrict__ xyz, const float* __restrict__ new_xyz,
    int* __restrict__ idx, int B, int N, int S, int ns, float r2) {
#pragma clang fp contract(off)
  const int q = blockIdx.x * blockDim.x + threadIdx.x;
  if (q >= B * S) return;
  const int b = q / S;
  const float ax = new_xyz[q * 3 + 0];
  const float ay = new_xyz[q * 3 + 1];
  const float az = new_xyz[q * 3 + 2];
  const float aa = (ax * ax + ay * ay) + az * az;
  const float* X = xyz + (size_t)b * N * 3;
  int* o = idx + (size_t)q * ns;
  int cnt = 0, first = -1;
  for (int j = 0; j < N && cnt < ns; ++j) {
    float bx = X[j * 3 + 0], by = X[j * 3 + 1], bz = X[j * 3 + 2];
    float bb = (bx * bx + by * by) + bz * bz;
    float ab = (ax * bx + ay * by) + az * bz;
    float sq = (aa - 2.0f * ab) + bb;
    if (!(sq > r2)) { if (cnt == 0) first = j; o[cnt++] = j; }
  }
  if (cnt == 0) { for (int k = 0; k < ns; ++k) o[k] = N - 1; }
  else          { for (int k = cnt; k < ns; ++k) o[k] = first; }
}

// ---------------------------------------------------------------------------
// Fused per-group MLP via V_WMMA_F32_16X16X4_F32.
// A-frag (16x4 f32): lane m=l&15, VGPR v holds K = kk + 2*(l>>4) + v.
// B-frag (4x16):     lane n=l&15, VGPR v holds K = kk + 2*(l>>4) + v.
// C/D (16x16):       lane n=l&15, VGPR v holds M = v + 8*(l>>4).
// Padded-K columns have A==0 in LDS, so the weight index is merely CLAMPED
// (branchless, finite garbage * 0 == 0) -- keeps EXEC all-ones at the WMMA.
// ---------------------------------------------------------------------------
__device__ __forceinline__ void wmma_layer(
    const float* __restrict__ W, const float* __restrict__ bias,
    const float* __restrict__ g, const float* __restrict__ be,
    const float* __restrict__ bufIn, int sIn, int cin, int cinp,
    float* __restrict__ bufOut, int sOut, int cout,
    int wave, int l15, int lhi) {
  for (int t = wave; t < cout / 16; t += 8) {
    const int n = t * 16 + l15;
    const float* Wr = W + (size_t)n * cin;
    v8f acc = {};
    for (int kk = 0; kk < cinp; kk += 4) {
      const int k0 = kk + 2 * lhi;
      v2f a, bf;
      a[0] = bufIn[l15 * sIn + k0];
      a[1] = bufIn[l15 * sIn + k0 + 1];
      bf[0] = Wr[min(k0,     cin - 1)];
      bf[1] = Wr[min(k0 + 1, cin - 1)];
      acc = __builtin_amdgcn_wmma_f32_16x16x4_f32(false, a, false, bf, (short)0,
                                                  acc, false, false);
    }
    const float sc = g[n], bb = bias[n], bt = be[n];
#pragma unroll
    for (int v = 0; v < 8; ++v) {
      const int m = v + 8 * lhi;
      float y = sc * ((acc[v] + bb) * INV_SQ) + bt;
      bufOut[m * sOut + n] = fmaxf(y, 0.0f);
    }
  }
}

__device__ __forceinline__ void wmma_layer_final(
    const float* __restrict__ W, const float* __restrict__ bias,
    const float* __restrict__ g, const float* __restrict__ be,
    const float* __restrict__ bufIn, int sIn, int cin,
    unsigned int* __restrict__ out, int cout,
    int wave, int l15, int lhi) {
  for (int t = wave; t < cout / 16; t += 8) {
    const int n = t * 16 + l15;
    const float* Wr = W + (size_t)n * cin;
    v8f acc = {};
    for (int kk = 0; kk < cin; kk += 4) {   // cin is a multiple of 16 here
      const int k0 = kk + 2 * lhi;
      v2f a, bf;
      a[0] = bufIn[l15 * sIn + k0];
      a[1] = bufIn[l15 * sIn + k0 + 1];
      bf[0] = Wr[k0];
      bf[1] = Wr[k0 + 1];
      acc = __builtin_amdgcn_wmma_f32_16x16x4_f32(false, a, false, bf, (short)0,
                                                  acc, false, false);
    }
    const float sc = g[n], bb = bias[n], bt = be[n];
    float mx = 0.0f;   // post-ReLU values are >= 0
#pragma unroll
    for (int v = 0; v < 8; ++v) {
      float y = sc * ((acc[v] + bb) * INV_SQ) + bt;
      mx = fmaxf(mx, fmaxf(y, 0.0f));
    }
    mx = fmaxf(mx, __shfl_xor(mx, 16, 32));  // combine M=0..7 with M=8..15
    if (lhi == 0) atomicMax(&out[n], __float_as_uint(mx));  // monotone for f>=0
  }
}

// One block per group (b,s). Gathers K samples x CIN features into LDS,
// runs layer1/layer2 ping-pong in LDS, streams layer3 through max-pool.
template <int K, int CIN, int C1, int C2, int C3, int WA, int WB>
__global__ void __launch_bounds__(256) sa_mlp_max_kernel(
    const float* __restrict__ xyz, const float* __restrict__ points,
    const float* __restrict__ new_xyz, const int* __restrict__ idx,
    const float* __restrict__ w1, const float* __restrict__ b1,
    const float* __restrict__ g1, const float* __restrict__ be1,
    const float* __restrict__ w2, const float* __restrict__ b2,
    const float* __restrict__ g2, const float* __restrict__ be2,
    const float* __restrict__ w3, const float* __restrict__ b3,
    const float* __restrict__ g3, const float* __restrict__ be3,
    float* __restrict__ out, int N, int S) {
  extern __shared__ float smem[];
  float* bufA = smem;            // 16 x WA  (input / layer2 output)
  float* bufB = smem + 16 * WA;  // 16 x WB  (layer1 output)
  constexpr int CINP = (CIN + 3) & ~3;
  constexpr int CP = CIN - 3;
  const int grp = blockIdx.x;
  const int b = grp / S;
  const int tid = threadIdx.x;
  const int lane = tid & 31, wave = tid >> 5;
  const int l15 = lane & 15, lhi = lane >> 4;
  unsigned int* og = (unsigned int*)(out + (size_t)grp * C3);

  for (int chunk = 0; chunk < K / 16; ++chunk) {
    // Gather 16 samples into LDS (zero-pad channels CIN..CINP-1)
    for (int e = tid; e < 16 * CINP; e += 256) {
      const int r = e / CINP, c = e % CINP;
      const int j = idx[(size_t)grp * K + chunk * 16 + r];
      float v = 0.0f;
      if (c < 3)        v = xyz[((size_t)b * N + j) * 3 + c] - new_xyz[(size_t)grp * 3 + c];
      else if (c < CIN) v = points[((size_t)b * N + j) * CP + (c - 3)];
      bufA[r * WA + c] = v;
    }
    __syncthreads();
    wmma_layer(w1, b1, g1, be1, bufA, WA, CIN, CINP, bufB, WB, C1, wave, l15, lhi);
    __syncthreads();
    wmma_layer(w2, b2, g2, be2, bufB, WB, C1, C1, bufA, WA, C2, wave, l15, lhi);
    __syncthreads();
    wmma_layer_final(w3, b3, g3, be3, bufA, WA, C2, og, C3, wave, l15, lhi);
    __syncthreads();
  }
}

// ---------------------------------------------------------------------------
// SA4 helper: identity sample indices + zero "new_xyz" (group_all path).
// ---------------------------------------------------------------------------
__global__ void init_sa4_kernel(int* __restrict__ idx4, float* __restrict__ nx4) {
  const int t = blockIdx.x * blockDim.x + threadIdx.x;
  if (t < 8 * 64) idx4[t] = t % 64;
  if (t < 8 * 3) nx4[t] = 0.0f;
}

// ---------------------------------------------------------------------------
// FC head: (8,1024) -> 512 -> 256 -> 24, single block, intermediates in LDS.
// ---------------------------------------------------------------------------
__global__ void __launch_bounds__(256) fc_kernel(
    const float* __restrict__ p4,
    const float* __restrict__ w1, const float* __restrict__ b1,
    const float* __restrict__ g1, const float* __restrict__ be1,
    const float* __restrict__ w2, const float* __restrict__ b2,
    const float* __restrict__ g2, const float* __restrict__ be2,
    const float* __restrict__ w3, const float* __restrict__ b3,
    float* __restrict__ out) {
  __shared__ float x[8 * 1024];
  __shared__ float h1[8 * 512];
  __shared__ float h2[8 * 256];
  const int tid = threadIdx.x;
  for (int i = tid; i < 8 * 1024; i += 256) x[i] = p4[i];
  __syncthreads();
  for (int o = tid; o < 8 * 512; o += 256) {
    const int r = o / 512, c = o % 512;
    float acc = 0.0f;
    const float* w = w1 + (size_t)c * 1024;
    const float* xr = x + r * 1024;
    for (int k = 0; k < 1024; ++k) acc += xr[k] * w[k];
    float y = g1[c] * ((acc + b1[c]) * INV_SQ) + be1[c];
    h1[o] = fmaxf(y, 0.0f);
  }
  __syncthreads();
  for (int o = tid; o < 8 * 256; o += 256) {
    const int r = o / 256, c = o % 256;
    float acc = 0.0f;
    const float* w = w2 + (size_t)c * 512;
    const float* xr = h1 + r * 512;
    for (int k = 0; k < 512; ++k) acc += xr[k] * w[k];
    float y = g2[c] * ((acc + b2[c]) * INV_SQ) + be2[c];
    h2[o] = fmaxf(y, 0.0f);
  }
  __syncthreads();
  for (int o = tid; o < 8 * 24; o += 256) {
    const int r = o / 24, c = o % 24;
    float acc = 0.0f;
    const float* w = w3 + (size_t)c * 256;
    const float* xr = h2 + r * 256;
    for (int k = 0; k < 256; ++k) acc += xr[k] * w[k];
    out[o] = acc + b3[c];
  }
}

// ---------------------------------------------------------------------------
extern "C" void kernel_launch(void* const* d_in, const int* in_sizes, int n_in,
                              void* d_out, int out_size, void* d_ws, size_t ws_size,
                              hipStream_t stream) {
  (void)out_size; (void)ws_size;
  const int B = 8, N0 = 8192;
  // Resolve input ordering from element counts (dict insertion vs pytree-sorted).
  const float *scene, *inst;
  int pbase;
  if (in_sizes[0] == B * N0 * 3) {
    scene = (const float*)d_in[0]; inst = (const float*)d_in[1]; pbase = 2;
  } else {
    inst = (const float*)d_in[0]; scene = (const float*)d_in[n_in - 1]; pbase = 1;
  }
  int saBase, fcBase;
  if (in_sizes[pbase] == 192) { saBase = pbase; fcBase = pbase + 48; }  // sa1 w1 = 64x3
  else                        { fcBase = pbase; saBase = pbase + 10; }  // fc-first (sorted)
  auto SA = [&](int s, int l, int t) { return (const float*)d_in[saBase + s * 12 + l * 4 + t]; };
  auto FC = [&](int i)               { return (const float*)d_in[fcBase + i]; };

  // Workspace carve-out (~9 MB total).
  char* ws = (char*)d_ws;
  size_t off = 0;
  auto alloc = [&](size_t bytes) {
    size_t cur = off; off += (bytes + 255) & ~(size_t)255; return (void*)(ws + cur);
  };
  float* nx1 = (float*)alloc((size_t)B * 1024 * 3 * 4);
  int*   id1 = (int*)  alloc((size_t)B * 1024 * 32 * 4);
  float* p1  = (float*)alloc((size_t)B * 1024 * 128 * 4);
  float* nx2 = (float*)alloc((size_t)B * 256 * 3 * 4);
  int*   id2 = (int*)  alloc((size_t)B * 256 * 48 * 4);
  float* p2  = (float*)alloc((size_t)B * 256 * 256 * 4);
  float* nx3 = (float*)alloc((size_t)B * 64 * 3 * 4);
  int*   id3 = (int*)  alloc((size_t)B * 64 * 64 * 4);
  float* p3  = (float*)alloc((size_t)B * 64 * 512 * 4);
  float* nx4 = (float*)alloc((size_t)B * 3 * 4);
  int*   id4 = (int*)  alloc((size_t)B * 64 * 4);
  float* p4  = (float*)alloc((size_t)B * 1024 * 4);

  // Max-pool targets must start at +0.0f (uint 0) for the atomicMax trick.
  hipMemsetAsync(p1, 0, (size_t)B * 1024 * 128 * 4, stream);
  hipMemsetAsync(p2, 0, (size_t)B * 256 * 256 * 4, stream);
  hipMemsetAsync(p3, 0, (size_t)B * 64 * 512 * 4, stream);
  hipMemsetAsync(p4, 0, (size_t)B * 1024 * 4, stream);
  init_sa4_kernel<<<2, 256, 0, stream>>>(id4, nx4);

  // ---- SA1: anchor=instance, npoint=1024, r=0.1, ns=32, 3->64->64->128
  fps_kernel<<<B, 256, 0, stream>>>(inst, nx1, 1024, 1024);
  ball_query_kernel<<<(B * 1024 + 255) / 256, 256, 0, stream>>>(
      scene, nx1, id1, B, N0, 1024, 32, (float)(0.1 * 0.1));
  sa_mlp_max_kernel<32, 3, 64, 64, 128, 64, 64>
      <<<B * 1024, 256, 16 * (64 + 64) * 4, stream>>>(
          scene, nullptr, nx1, id1,
          SA(0,0,0), SA(0,0,1), SA(0,0,2), SA(0,0,3),
          SA(0,1,0), SA(0,1,1), SA(0,1,2), SA(0,1,3),
          SA(0,2,0), SA(0,2,1), SA(0,2,2), SA(0,2,3),
          p1, N0, 1024);

  // ---- SA2: npoint=256, r=0.2, ns=48, 131->128->128->256
  fps_kernel<<<B, 256, 0, stream>>>(nx1, nx2, 1024, 256);
  ball_query_kernel<<<(B * 256 + 255) / 256, 256, 0, stream>>>(
      nx1, nx2, id2, B, 1024, 256, 48, (float)(0.2 * 0.2));
  sa_mlp_max_kernel<48, 131, 128, 128, 256, 132, 128>
      <<<B * 256, 256, 16 * (132 + 128) * 4, stream>>>(
          nx1, p1, nx2, id2,
          SA(1,0,0), SA(1,0,1), SA(1,0,2), SA(1,0,3),
          SA(1,1,0), SA(1,1,1), SA(1,1,2), SA(1,1,3),
          SA(1,2,0), SA(1,2,1), SA(1,2,2), SA(1,2,3),
          p2, 1024, 256);

  // ---- SA3: npoint=64, r=0.4, ns=64, 259->256->256->512
  fps_kernel<<<B, 256, 0, stream>>>(nx2, nx3, 256, 64);
  ball_query_kernel<<<(B * 64 + 255) / 256, 256, 0, stream>>>(
      nx2, nx3, id3, B, 256, 64, 64, (float)(0.4 * 0.4));
  sa_mlp_max_kernel<64, 259, 256, 256, 512, 260, 256>
      <<<B * 64, 256, 16 * (260 + 256) * 4, stream>>>(
          nx2, p2, nx3, id3,
          SA(2,0,0), SA(2,0,1), SA(2,0,2), SA(2,0,3),
          SA(2,1,0), SA(2,1,1), SA(2,1,2), SA(2,1,3),
          SA(2,2,0), SA(2,2,1), SA(2,2,2), SA(2,2,3),
          p3, 256, 64);

  // ---- SA4: group_all (identity idx, zero new_xyz), 515->512->512->1024
  sa_mlp_max_kernel<64, 515, 512, 512, 1024, 516, 512>
      <<<B, 256, 16 * (516 + 512) * 4, stream>>>(
          nx3, p3, nx4, id4,
          SA(3,0,0), SA(3,0,1), SA(3,0,2), SA(3,0,3),
          SA(3,1,0), SA(3,1,1), SA(3,1,2), SA(3,1,3),
          SA(3,2,0), SA(3,2,1), SA(3,2,2), SA(3,2,3),
          p4, 64, 1);

  // ---- FC head -> (64, 3) flat = 192 floats
  fc_kernel<<<1, 256, 0, stream>>>(p4,
      FC(0), FC(1), FC(2), FC(3),
      FC(4), FC(5), FC(6), FC(7),
      FC(8), FC(9), (float*)d_out);
}